// AvgPool2d_4217657885202
// MI455X (gfx1250) — compile-verified
//
#include <hip/hip_runtime.h>
#include <cstdint>

// Problem constants (from reference: x = (8, 64, 512, 512) f32, kernel 64x64)
#define N_   8
#define C_   64
#define H_   512
#define W_   512
#define K_   64
#define HV_  (H_ - K_ + 1)     // 449 valid rows/cols
#define PT_  31                // top/left replicate pad ((512-449)//2)
#define P_   (N_ * C_)         // 512 planes
#define INV_AREA (1.0f / 4096.0f)

// ---------------------------------------------------------------------------
// Pass 1: vertical sliding-window sum (window K_ rows), per plane.
// Thread owns one column x. Software-pipelined: 8 lead + 8 trail loads per
// chunk issue as a clause (16 outstanding loads/wave) before the serial
// running-sum chain consumes them from registers.
// V[p][r][x] = sum_{y=r..r+63} X[p][y][x],  r in [0,449)
// ---------------------------------------------------------------------------
__global__ __launch_bounds__(256) void avgpool_vpass(
    const float* __restrict__ X, float* __restrict__ V) {
  const int x = blockIdx.x * 256 + threadIdx.x;   // 0..511
  const int p = blockIdx.y;                        // 0..511
  const float* xp = X + (size_t)p * H_ * W_ + x;
  float*       vp = V + (size_t)p * HV_ * W_ + x;

  // init: sum rows 0..62 with 4 independent accumulators (loads clause up)
  float s0 = 0.f, s1 = 0.f, s2 = 0.f, s3 = 0.f;
#pragma unroll
  for (int y = 0; y < 60; y += 4) {
    s0 += xp[(size_t)(y + 0) * W_];
    s1 += xp[(size_t)(y + 1) * W_];
    s2 += xp[(size_t)(y + 2) * W_];
    s3 += xp[(size_t)(y + 3) * W_];
  }
  float s = (s0 + s1) + (s2 + s3)
          + xp[(size_t)60 * W_] + xp[(size_t)61 * W_] + xp[(size_t)62 * W_];

  // main: 56 chunks of 8 cover r = 0..447
  for (int rb = 0; rb < HV_ - 1; rb += 8) {
    float lead[8], trail[8];
#pragma unroll
    for (int i = 0; i < 8; ++i)
      lead[i] = xp[(size_t)(rb + i + K_ - 1) * W_];          // rows rb+63..rb+70
#pragma unroll
    for (int i = 0; i < 8; ++i)
      trail[i] = __builtin_nontemporal_load(xp + (size_t)(rb + i) * W_); // last use
    int pfr = rb + K_ - 1 + 32; pfr = pfr < H_ - 1 ? pfr : H_ - 1;
    __builtin_prefetch(xp + (size_t)pfr * W_, 0, 0);         // global_prefetch_b8
#pragma unroll
    for (int i = 0; i < 8; ++i) {
      s += lead[i];
      vp[(size_t)(rb + i) * W_] = s;
      s -= trail[i];
    }
  }
  // remainder: r = 448
  s += xp[(size_t)(HV_ - 1 + K_ - 1) * W_];                  // row 511
  vp[(size_t)(HV_ - 1) * W_] = s;
}

// ---------------------------------------------------------------------------
// Pass 2: horizontal sliding-window sum + divide + replicate-pad to 512x512.
// Thread owns one FINAL output row y (row replication handled by clamping r).
// Window init (first 63 V values) staged into LDS with async-to-LDS B128
// (ASYNCcnt path). Main loop software-pipelined in chunks of 8; the 8 lead
// values are contiguous & 16B-aligned -> two global_load_b128 per chunk.
// ---------------------------------------------------------------------------
#define RSTRIDE 68   // floats per thread in LDS ring: 68*4 = 272 B (16B aligned, bank-spread)

__global__ __launch_bounds__(128) void avgpool_hpass(
    const float* __restrict__ V, float* __restrict__ out) {
  __shared__ float ring[128 * RSTRIDE];   // 34,816 B

  const int tid = threadIdx.x;
  const int y   = blockIdx.x * 128 + tid;  // 0..511 final output row
  const int p   = blockIdx.y;              // plane

  int r = y - PT_;
  r = r < 0 ? 0 : (r > HV_ - 1 ? HV_ - 1 : r);          // replicate rows

  const float* vp = V   + (size_t)p * HV_ * W_ + (size_t)r * W_;
  float*       op = out + (size_t)p * H_  * W_ + (size_t)y * W_;

  // --- async-stage first 64 floats of this thread's V row into LDS ---------
  const uint32_t lds = (uint32_t)(uintptr_t)(&ring[tid * RSTRIDE]); // low 32 bits = LDS offset
  const uint64_t ga  = (uint64_t)(uintptr_t)vp;
#pragma unroll
  for (int k = 0; k < 16; ++k) {
    uint32_t l = lds + (uint32_t)(k * 16);
    uint64_t g = ga  + (uint64_t)(k * 16);
    asm volatile("global_load_async_to_lds_b128 %0, %1, off"
                 :: "v"(l), "v"(g) : "memory");
  }
  asm volatile("s_wait_asynccnt 0" ::: "memory");        // ASYNCcnt == 0

  // init window sum over j = 0..62 from LDS (ds_load_b128 x15 + 3 scalars)
  float s = 0.0f;
  const float4* r4 = (const float4*)(&ring[tid * RSTRIDE]);
#pragma unroll
  for (int k = 0; k < 15; ++k) { float4 v = r4[k]; s += (v.x + v.y) + (v.z + v.w); }
  s += ring[tid * RSTRIDE + 60] + ring[tid * RSTRIDE + 61] + ring[tid * RSTRIDE + 62];

  // --- c = 0 (peeled, left replicate) --------------------------------------
  s += vp[K_ - 1];
  const float first = s * INV_AREA;
  op[PT_] = first;
  s -= __builtin_nontemporal_load(vp + 0);

  // --- main: 55 chunks of 8 cover c = 1..440 -------------------------------
  for (int cb = 1; cb < 434; cb += 8) {
    // lead: vp[cb+63 .. cb+70] contiguous, (cb+63) = 64+8k -> 16B aligned
    const float4 L0 = *(const float4*)(vp + cb + K_ - 1);
    const float4 L1 = *(const float4*)(vp + cb + K_ + 3);
    float lead[8] = { L0.x, L0.y, L0.z, L0.w, L1.x, L1.y, L1.z, L1.w };
    float trail[8];
#pragma unroll
    for (int i = 0; i < 8; ++i)
      trail[i] = __builtin_nontemporal_load(vp + cb + i);  // last use
    int pfc = cb + K_ - 1 + 64; pfc = pfc < W_ - 1 ? pfc : W_ - 1;
    __builtin_prefetch(vp + pfc, 0, 0);                    // global_prefetch_b8
#pragma unroll
    for (int i = 0; i < 8; ++i) {
      s += lead[i];
      op[cb + i + PT_] = s * INV_AREA;
      s -= trail[i];
    }
  }

  // --- tail: c = 441..447 --------------------------------------------------
  for (int c = 441; c < HV_ - 1; ++c) {
    s += vp[c + K_ - 1];
    op[c + PT_] = s * INV_AREA;
    s -= __builtin_nontemporal_load(vp + c);
  }

  // --- c = 448 (peeled, right replicate) -----------------------------------
  s += vp[W_ - 1];
  const float last = s * INV_AREA;
  op[HV_ - 1 + PT_] = last;                              // op[479]

  for (int xx = 0; xx < PT_; ++xx)        op[xx] = first;  // x = 0..30
  for (int xx = PT_ + HV_; xx < W_; ++xx) op[xx] = last;   // x = 480..511
}

// ---------------------------------------------------------------------------
// Fallback (only if workspace is too small): direct 64x64 summation.
// Slow but correct and deterministic.
// ---------------------------------------------------------------------------
__global__ __launch_bounds__(256) void avgpool_direct(
    const float* __restrict__ X, float* __restrict__ out) {
  const size_t i = (size_t)blockIdx.x * 256 + threadIdx.x;
  const size_t total = (size_t)P_ * H_ * W_;
  if (i >= total) return;
  const int xcol = (int)(i % W_);
  const int yrow = (int)((i / W_) % H_);
  const size_t p = i / ((size_t)H_ * W_);
  int r = yrow - PT_; r = r < 0 ? 0 : (r > HV_ - 1 ? HV_ - 1 : r);
  int c = xcol - PT_; c = c < 0 ? 0 : (c > HV_ - 1 ? HV_ - 1 : c);
  const float* xp = X + p * (size_t)H_ * W_;
  float s = 0.0f;
  for (int a = 0; a < K_; ++a) {
    const float* row = xp + (size_t)(r + a) * W_ + c;
    for (int b = 0; b < K_; ++b) s += row[b];
  }
  out[i] = s * INV_AREA;
}

// ---------------------------------------------------------------------------
extern "C" void kernel_launch(void* const* d_in, const int* in_sizes, int n_in,
                              void* d_out, int out_size, void* d_ws, size_t ws_size,
                              hipStream_t stream) {
  (void)in_sizes; (void)n_in; (void)out_size;
  const float* X   = (const float*)d_in[0];
  float*       out = (float*)d_out;

  const size_t needV = (size_t)P_ * HV_ * W_ * sizeof(float); // ~449 MB
  if (d_ws != nullptr && ws_size >= needV) {
    float* V = (float*)d_ws;
    avgpool_vpass<<<dim3(W_ / 256, P_), 256, 0, stream>>>(X, V);
    avgpool_hpass<<<dim3(H_ / 128, P_), 128, 0, stream>>>(V, out);
  } else {
    const size_t total = (size_t)P_ * H_ * W_;
    avgpool_direct<<<(unsigned)((total + 255) / 256), 256, 0, stream>>>(X, out);
  }
}